// DualEdgeModel_90950227460162
// MI455X (gfx1250) — compile-verified
//
#include <hip/hip_runtime.h>
#include <hip/hip_bf16.h>

typedef float v2f __attribute__((ext_vector_type(2)));
typedef float v8f __attribute__((ext_vector_type(8)));

// ---------------------------------------------------------------------------
// Kernel 1: zero the per-node accumulator T [N*6]
// ---------------------------------------------------------------------------
__global__ void k_zero(float* __restrict__ T, int n) {
    int i = blockIdx.x * blockDim.x + threadIdx.x;
    if (i < n) T[i] = 0.0f;
}

// ---------------------------------------------------------------------------
// Kernel 2: build edge features x = [feat[src], feat[dst]] and scatter-add
// into node sums T[n] = sum over edges incident to n of x_e  (S-matrix sums)
// ---------------------------------------------------------------------------
__global__ void k_scatter(const float* __restrict__ node_feats,
                          const int* __restrict__ src,
                          const int* __restrict__ dst,
                          float* __restrict__ xfeat,
                          float* __restrict__ T,
                          int E) {
    int i = blockIdx.x * blockDim.x + threadIdx.x;
    if (i >= E) return;
    int s = src[i];
    int d = dst[i];
    float x[6];
#pragma unroll
    for (int c = 0; c < 3; ++c) {
        x[c]     = node_feats[s * 3 + c];
        x[3 + c] = node_feats[d * 3 + c];
    }
#pragma unroll
    for (int c = 0; c < 6; ++c) xfeat[i * 6 + c] = x[c];
#pragma unroll
    for (int c = 0; c < 6; ++c) atomicAdd(&T[s * 6 + c], x[c]);
    if (d != s) {
#pragma unroll
        for (int c = 0; c < 6; ++c) atomicAdd(&T[d * 6 + c], x[c]);
    }
}

// ---------------------------------------------------------------------------
// Kernel 3: GIN aggregation using the identity
//   A@x_i = T[src_i] + [src!=dst]*T[dst_i] - x_i - P_i
// where P_i corrects for parallel edges (count_ij == 2), found by a tiled
// brute-force scan over all edge pairs through LDS.
//   h0_i = (1+eps)*x_i + A@x_i
// ---------------------------------------------------------------------------
__global__ __launch_bounds__(256) void k_aggregate(const float* __restrict__ xfeat,
                                                   const float* __restrict__ T,
                                                   const int* __restrict__ src,
                                                   const int* __restrict__ dst,
                                                   const float* __restrict__ eps_p,
                                                   float* __restrict__ h0,
                                                   int E) {
    __shared__ int s_s[256];
    __shared__ int s_d[256];

    int tid = threadIdx.x;
    int i = blockIdx.x * 256 + tid;
    bool active = (i < E);

    int si = 0, di = 0;
    float acc[6] = {0, 0, 0, 0, 0, 0};
    float xi[6]  = {0, 0, 0, 0, 0, 0};

    if (active) {
        si = src[i];
        di = dst[i];
        float onePlusEps = 1.0f + eps_p[0];
#pragma unroll
        for (int c = 0; c < 6; ++c) {
            xi[c]  = xfeat[i * 6 + c];
            acc[c] = onePlusEps * xi[c] + T[si * 6 + c] - xi[c];
        }
        if (si != di) {
#pragma unroll
            for (int c = 0; c < 6; ++c) acc[c] += T[di * 6 + c];
        }
    }

    bool scan = active && (si != di);   // P_i == 0 for self-loop edges
    int ntiles = (E + 255) / 256;
    for (int t = 0; t < ntiles; ++t) {
        int j = t * 256 + tid;
        s_s[tid] = (j < E) ? src[j] : -1;
        s_d[tid] = (j < E) ? dst[j] : -1;
        __syncthreads();
        if (scan) {
            for (int k = 0; k < 256; ++k) {
                int js = s_s[k];
                int jd = s_d[k];
                if ((js == si && jd == di) || (js == di && jd == si)) {
                    int e = t * 256 + k;
#pragma unroll
                    for (int c = 0; c < 6; ++c) acc[c] -= xfeat[e * 6 + c];
                }
            }
        }
        __syncthreads();
    }

    if (active) {
#pragma unroll
        for (int c = 0; c < 6; ++c) h0[i * 6 + c] = acc[c];
    }
}

// ---------------------------------------------------------------------------
// Kernel 4: fused 3-layer MLP. 256 threads = 8 wave32, 32 rows per block.
// Each wave owns one 16x16 C tile (rt = wave/4 row-tile, ct = wave%4 col-tile).
// Layer1: K=6 padded to 8 -> 2x V_WMMA_F32_16X16X4_F32
// Layer2: K=64           -> 16x V_WMMA_F32_16X16X4_F32
// Layer3: 64-wide dot per row (VALU).
// All activations stay in LDS (stride 66 padding avoids bank conflicts).
// ---------------------------------------------------------------------------
#define HSTRIDE 66

__global__ __launch_bounds__(256) void k_mlp(const float* __restrict__ h0,
                                             const float* __restrict__ W1,
                                             const float* __restrict__ b1,
                                             const float* __restrict__ W2,
                                             const float* __restrict__ b2,
                                             const float* __restrict__ W3,
                                             const float* __restrict__ b3,
                                             float* __restrict__ out,
                                             int E) {
    __shared__ float sW1[8 * 64];          // K padded 6 -> 8
    __shared__ float sW2[64 * 64];
    __shared__ float sb1[64];
    __shared__ float sb2[64];
    __shared__ float sW3[64];
    __shared__ float sb3;
    __shared__ float sX[32 * 10];          // input rows, K padded to 8 (stride 10)
    __shared__ float sH1[32 * HSTRIDE];    // layer-1 activations
    __shared__ float sH2[32 * HSTRIDE];    // layer-2 activations

    int tid = threadIdx.x;
    int rowbase = blockIdx.x * 32;

    // Stage weights/biases in LDS.
    for (int idx = tid; idx < 8 * 64; idx += 256) {
        int k = idx >> 6, c = idx & 63;
        sW1[idx] = (k < 6) ? W1[k * 64 + c] : 0.0f;
    }
    for (int idx = tid; idx < 64 * 64; idx += 256) sW2[idx] = W2[idx];
    if (tid < 64) {
        sb1[tid] = b1[tid];
        sb2[tid] = b2[tid];
        sW3[tid] = W3[tid];
    }
    if (tid == 0) sb3 = b3[0];

    // Stage this block's 32 input rows (K padded with zeros).
    for (int idx = tid; idx < 32 * 8; idx += 256) {
        int r = idx >> 3, k = idx & 7;
        int row = rowbase + r;
        sX[r * 10 + k] = (k < 6 && row < E) ? h0[row * 6 + k] : 0.0f;
    }
    __syncthreads();

    int lane = tid & 31;
    int wave = tid >> 5;
    int rt = wave >> 2;          // 0..1  (16-row tile)
    int ct = wave & 3;           // 0..3  (16-col tile)
    int half = lane >> 4;        // K-half selector for A/B fragments
    int l16 = lane & 15;
    int rb = rt * 16;
    int cb = ct * 16;

    // ---- Layer 1: h1 = relu(X @ W1 + b1), K = 8 (padded) ----
    v8f c1;
    {
        float bias = sb1[cb + l16];
#pragma unroll
        for (int r = 0; r < 8; ++r) c1[r] = bias;
    }
#pragma unroll
    for (int k = 0; k < 8; k += 4) {
        v2f a, bv;
        int kk = k + 2 * half;
        a.x  = sX[(rb + l16) * 10 + kk];
        a.y  = sX[(rb + l16) * 10 + kk + 1];
        bv.x = sW1[kk * 64 + cb + l16];
        bv.y = sW1[(kk + 1) * 64 + cb + l16];
        c1 = __builtin_amdgcn_wmma_f32_16x16x4_f32(false, a, false, bv,
                                                   (short)0, c1, false, false);
    }
#pragma unroll
    for (int r = 0; r < 8; ++r) {
        float v = c1[r];
        v = v > 0.0f ? v : 0.0f;
        sH1[(rb + r + 8 * half) * HSTRIDE + cb + l16] = v;
    }
    __syncthreads();

    // ---- Layer 2: h2 = relu(h1 @ W2 + b2), K = 64 ----
    v8f c2;
    {
        float bias = sb2[cb + l16];
#pragma unroll
        for (int r = 0; r < 8; ++r) c2[r] = bias;
    }
#pragma unroll
    for (int k = 0; k < 64; k += 4) {
        v2f a, bv;
        int kk = k + 2 * half;
        a.x  = sH1[(rb + l16) * HSTRIDE + kk];
        a.y  = sH1[(rb + l16) * HSTRIDE + kk + 1];
        bv.x = sW2[kk * 64 + cb + l16];
        bv.y = sW2[(kk + 1) * 64 + cb + l16];
        c2 = __builtin_amdgcn_wmma_f32_16x16x4_f32(false, a, false, bv,
                                                   (short)0, c2, false, false);
    }
#pragma unroll
    for (int r = 0; r < 8; ++r) {
        float v = c2[r];
        v = v > 0.0f ? v : 0.0f;
        sH2[(rb + r + 8 * half) * HSTRIDE + cb + l16] = v;
    }
    __syncthreads();

    // ---- Layer 3: out = relu(h2 @ W3 + b3), per-row 64-dot ----
    if (tid < 32) {
        int row = rowbase + tid;
        if (row < E) {
            float s = sb3;
#pragma unroll
            for (int c = 0; c < 64; ++c) s += sH2[tid * HSTRIDE + c] * sW3[c];
            out[row] = s > 0.0f ? s : 0.0f;
        }
    }
}

// ---------------------------------------------------------------------------
extern "C" void kernel_launch(void* const* d_in, const int* in_sizes, int n_in,
                              void* d_out, int out_size, void* d_ws, size_t ws_size,
                              hipStream_t stream) {
    const float* node_feats = (const float*)d_in[0];
    const int*   edge_idx   = (const int*)d_in[1];
    const float* eps        = (const float*)d_in[2];
    const float* W1         = (const float*)d_in[3];
    const float* b1         = (const float*)d_in[4];
    const float* W2         = (const float*)d_in[5];
    const float* b2         = (const float*)d_in[6];
    const float* W3         = (const float*)d_in[7];
    const float* b3         = (const float*)d_in[8];
    float* out = (float*)d_out;

    int N = in_sizes[0] / 3;
    int E = in_sizes[1] / 2;
    const int* src = edge_idx;
    const int* dst = edge_idx + E;

    // Workspace layout: xfeat [E*6] | T [N*6] | h0 [E*6]
    float* xfeat = (float*)d_ws;
    float* T     = xfeat + (size_t)E * 6;
    float* h0    = T + (size_t)N * 6;

    k_zero<<<(N * 6 + 255) / 256, 256, 0, stream>>>(T, N * 6);
    k_scatter<<<(E + 255) / 256, 256, 0, stream>>>(node_feats, src, dst, xfeat, T, E);
    k_aggregate<<<(E + 255) / 256, 256, 0, stream>>>(xfeat, T, src, dst, eps, h0, E);
    k_mlp<<<(E + 31) / 32, 256, 0, stream>>>(h0, W1, b1, W2, b2, W3, b3, out, E);
}